// SparseNodeLinear_77799037600415
// MI455X (gfx1250) — compile-verified
//
#include <hip/hip_runtime.h>

typedef float v2f __attribute__((ext_vector_type(2)));
typedef float v4f __attribute__((ext_vector_type(4)));
typedef float v8f __attribute__((ext_vector_type(8)));

#define Bsz   8        // batch
#define Nn    2048     // nodes
#define Kk    2049     // N+1 (reduction length)
#define Ff    32       // out features
#define KPAD  2052     // Kk padded to multiple of 4
#define NG    513      // KPAD/4 k-groups
#define XSTR  17       // padded row stride (floats) for transposed x in LDS
#define GPW   65       // ceil(NG/8) groups per wave
#define CRED_FLOATS (8 * 2 * 8 * 32)   // 8 waves x 2 tiles x 8 vgprs x 32 lanes

__global__ __launch_bounds__(256)
void sparse_node_linear_wmma(const float* __restrict__ x,
                             const float* __restrict__ wgt,
                             const float* __restrict__ bias,
                             const unsigned char* __restrict__ emask,
                             float* __restrict__ out)
{
    extern __shared__ float lds[];
    float* xk   = lds;                   // [KPAD][XSTR]  masked x, transposed
    float* cred = lds + KPAD * XSTR;     // [8 waves][2 tiles][8 vgpr][32 lanes]

    const int n    = blockIdx.x;
    const int tid  = threadIdx.x;
    const int lane = tid & 31;
    // wave index is uniform per wave: pin it to an SGPR so loop bounds,
    // tail predicate and row bases are scalar (no exec-masked vector loops).
    const int wave = __builtin_amdgcn_readfirstlane(tid >> 5);

    // ---- Phase 0: zero xk (pad rows + batch slots 8..15). cred is fully
    // overwritten in Phase 3 before Phase 4 reads it, so skip it here.
    {
        v4f z4 = {0.f, 0.f, 0.f, 0.f};
        v4f* p = (v4f*)xk;
        const int total_q = (KPAD * XSTR) / 4;     // 34884/4 = 8721
        for (int i = tid; i < total_q; i += 256) p[i] = z4;   // ds_store_b128
    }
    __syncthreads();

    // ---- Phase 1: stage xm[b][k] = mask ? 0 : x[b][n][k], transposed into xk[k][b]
    {
        const float* xrow = x + ((size_t)wave * Nn + n) * Kk;   // scalar base
        const unsigned char* mrow = emask + (size_t)n * Kk;
        for (int k0 = 0; k0 < Kk; k0 += 32) {
            int k = k0 + lane;
            if (k < Kk) {
                float v = xrow[k];
                xk[k * XSTR + wave] = mrow[k] ? 0.0f : v;   // stride-17: conflict-free
            }
        }
    }
    __syncthreads();

    // ---- Phase 2: per-wave partial GEMM over its K-groups, fp32 WMMA 16x16x4
    // A fragment (16x4 f32): lanes 0-15 -> K=0/1 in v0/v1; lanes 16-31 -> K=2/3.  M = lane%16 (batch; rows 8..15 zero).
    // B fragment (4x16 f32): v0 lanes0-15 K=0, lanes16-31 K=2; v1 lanes0-15 K=1, lanes16-31 K=3.  N = lane%16 (feature).
    const int half = lane >> 4;          // 0: K pair {0,1}, 1: K pair {2,3}
    const int l16  = lane & 15;

    v8f c0 = {0.f, 0.f, 0.f, 0.f, 0.f, 0.f, 0.f, 0.f};  // features 0..15
    v8f c1 = c0;                                         // features 16..31

    const int g0   = wave * GPW;                          // scalar
    const int gcnt = (wave == 7) ? (NG - 1 - g0) : GPW;   // scalar trip count (57 or 65)

    const float* wp = wgt + (size_t)n * Kk * Ff
                          + (size_t)g0 * (4 * Ff)
                          + (size_t)(half * 2) * Ff + l16;
    int ai = g0 * (4 * XSTR) + half * (2 * XSTR) + l16;

    #pragma unroll 4
    for (int i = 0; i < gcnt; ++i) {
        v2f a, b0, b1;
        a.x  = xk[ai];
        a.y  = xk[ai + XSTR];
        b0.x = wp[0];          // imm offset   0
        b0.y = wp[Ff];         // imm offset 128
        b1.x = wp[16];         // imm offset  64
        b1.y = wp[Ff + 16];    // imm offset 192
        c0 = __builtin_amdgcn_wmma_f32_16x16x4_f32(false, a, false, b0, (short)0, c0, false, false);
        c1 = __builtin_amdgcn_wmma_f32_16x16x4_f32(false, a, false, b1, (short)0, c1, false, false);
        wp += 4 * Ff;
        ai += 4 * XSTR;
    }

    if (wave == 7) {
        // group NG-1: only k=2048 is a real row; xk rows 2049..2051 are zero,
        // so clamp all B-row addresses to row 2048 (their products are x0).
        v2f a, b0, b1;
        int at = (NG - 1) * (4 * XSTR) + half * (2 * XSTR) + l16;
        a.x = xk[at];
        a.y = xk[at + XSTR];
        const float* wlast = wgt + ((size_t)n * Kk + (Kk - 1)) * Ff;  // row k=2048
        b0.x = wlast[l16];
        b0.y = wlast[l16];
        b1.x = wlast[16 + l16];
        b1.y = wlast[16 + l16];
        c0 = __builtin_amdgcn_wmma_f32_16x16x4_f32(false, a, false, b0, (short)0, c0, false, false);
        c1 = __builtin_amdgcn_wmma_f32_16x16x4_f32(false, a, false, b1, (short)0, c1, false, false);
    }

    // ---- Phase 3: dump per-wave fragments to LDS (deterministic reduction)
    {
        float* cw = cred + wave * (2 * 8 * 32);
        #pragma unroll
        for (int v = 0; v < 8; ++v) {
            cw[v * 32 + lane]       = c0[v];
            cw[(8 + v) * 32 + lane] = c1[v];
        }
    }
    __syncthreads();

    // ---- Phase 4: reduce 8 wave-partials in fixed order, add bias, store.
    // wave w handles batch b=w; lane = feature f.  Fragment addr for (m=b<8, col=f):
    // tile t=f/16, vgpr v=b, frag-lane = f%16 (lanes 0-15 half).
    {
        const int t  = lane >> 4;
        const int fl = lane & 15;
        float s = 0.f;
        #pragma unroll
        for (int sw = 0; sw < 8; ++sw)
            s += cred[sw * (2 * 8 * 32) + (t * 8 + wave) * 32 + fl];
        s += bias[(size_t)n * Ff + lane];
        out[((size_t)wave * Nn + n) * Ff + lane] = s;   // coalesced 128B per wave
    }
}

extern "C" void kernel_launch(void* const* d_in, const int* in_sizes, int n_in,
                              void* d_out, int out_size, void* d_ws, size_t ws_size,
                              hipStream_t stream) {
    (void)in_sizes; (void)n_in; (void)out_size; (void)d_ws; (void)ws_size;
    const float*         xp = (const float*)d_in[0];          // [8,2048,2049] f32
    const float*         wp = (const float*)d_in[1];          // [2048,2049,32] f32
    const float*         bp = (const float*)d_in[2];          // [2048,32] f32
    const unsigned char* mp = (const unsigned char*)d_in[3];  // [2048,2049] bool
    float* op = (float*)d_out;                                // [8,2048,32] f32

    const size_t shmem = (size_t)(KPAD * XSTR + CRED_FLOATS) * sizeof(float); // ~152 KB
    (void)hipFuncSetAttribute((const void*)sparse_node_linear_wmma,
                              hipFuncAttributeMaxDynamicSharedMemorySize, (int)shmem);
    sparse_node_linear_wmma<<<dim3(Nn), dim3(256), shmem, stream>>>(xp, wp, bp, mp, op);
}